// IGNN_v2_60026462929134
// MI455X (gfx1250) — compile-verified
//
#include <hip/hip_runtime.h>
#include <hip/hip_bf16.h>

// IGNN_v2 forward on gfx1250 (MI455X), fp32 end-to-end.
//   A   = adj @ scaler_w^T + b        : WMMA GEMM (NT, bias)      4096x4096x16384
//   rho = power-iteration on |A|      : VALU matvec + reductions  (51 matvecs)
//   Wp  = l1-ball row projection of W : 1 block, per-row sort
//   B   = (omega1 @ features) @ A     : WMMA GEMM (NN) x2
//   50x : X = relu(Wp @ X @ A + B)    : WMMA GEMM (NN) + fused residual/relu
//   out = X^T @ V^T                   : tiny VALU kernel
//
// GEMM pipeline: double-buffered LDS tiles fed by CDNA5 async global->LDS
// copies (ASYNCcnt), 32 v_wmma_f32_16x16x4_f32 per K-step between barriers.

typedef __attribute__((ext_vector_type(2))) float v2f;
typedef __attribute__((ext_vector_type(8))) float v8f;

typedef int vi2 __attribute__((vector_size(8)));
typedef int vi4 __attribute__((vector_size(16)));
typedef __attribute__((address_space(1))) vi2 gas_vi2;
typedef __attribute__((address_space(3))) vi2 las_vi2;
typedef __attribute__((address_space(1))) vi4 gas_vi4;
typedef __attribute__((address_space(3))) vi4 las_vi4;

#define NHID 64
#define NNODE 4096
#define NCLASS 16

__device__ __forceinline__ void async_ld_b64(const void* g, void* l) {
#if __has_builtin(__builtin_amdgcn_global_load_async_to_lds_b64)
  __builtin_amdgcn_global_load_async_to_lds_b64(
      (gas_vi2*)(vi2*)g, (las_vi2*)(vi2*)l, 0, 0);
#else
  unsigned loff = (unsigned)(uintptr_t)(las_vi2*)(vi2*)l;
  asm volatile("global_load_async_to_lds_b64 %0, %1, off"
               :: "v"(loff), "v"(g) : "memory");
#endif
}

__device__ __forceinline__ void async_ld_b128(const void* g, void* l) {
#if __has_builtin(__builtin_amdgcn_global_load_async_to_lds_b128)
  __builtin_amdgcn_global_load_async_to_lds_b128(
      (gas_vi4*)(vi4*)g, (las_vi4*)(vi4*)l, 0, 0);
#else
  unsigned loff = (unsigned)(uintptr_t)(las_vi4*)(vi4*)l;
  asm volatile("global_load_async_to_lds_b128 %0, %1, off"
               :: "v"(loff), "v"(g) : "memory");
#endif
}

__device__ __forceinline__ void wait_async0() {
#if __has_builtin(__builtin_amdgcn_s_wait_asynccnt)
  __builtin_amdgcn_s_wait_asynccnt(0);
#else
  asm volatile("s_wait_asynccnt 0x0" ::: "memory");
#endif
}

// ---------------------------------------------------------------------------
// Generic fp32 WMMA GEMM:  C[M,N] = A[M,K] * opB + (bias) + (Res, relu)
//   BT=false : B is [K,N] row-major (NN)
//   BT=true  : B is [N,K] row-major, used transposed (NT); staged
//              un-transposed in LDS, transposed at fragment-read time.
// Block: 128 threads (4 wave32). Block tile 64(M) x 64(N), K-step 32,
// double-buffered LDS fed by async global->LDS copies.
// Wave w computes rows [w*16, w*16+16) x all 64 cols -> 4 v8f accumulators.
// ---------------------------------------------------------------------------
template <bool BT, bool BIAS, bool RESRELU>
__global__ __launch_bounds__(128) void gemm_f32_wmma(
    const float* __restrict__ A, const float* __restrict__ B,
    const float* __restrict__ bias, const float* __restrict__ Res,
    float* __restrict__ C, int M, int N, int K) {
  // A tile: 64 rows x 32 k, rows padded to 36 floats (144B: 8B-aligned rows,
  //         lane-half k-offsets land in disjoint bank sets).
  // B tile (NN): 32 k x 64 n, rows padded to 72 floats (288B: 16B-aligned,
  //         lane halves 16 banks apart -> conflict-free).
  // B tile (BT): 64 n x 32 k staged row-major like A (stride 36).
  __shared__ __align__(16) float As[2][64][36];
  __shared__ __align__(16) float Bsh[2][2304];  // max(32*72, 64*36)

  const int t    = threadIdx.x;
  const int lane = t & 31;
  const int wv   = t >> 5;     // wave id 0..3
  const int lh   = lane >> 4;  // lane half
  const int lid  = lane & 15;
  const int mBlock = blockIdx.y * 64;
  const int nBlock = blockIdx.x * 64;
  const int mBase  = wv * 16;

  v8f acc[4];
#pragma unroll
  for (int i = 0; i < 4; ++i) acc[i] = (v8f){};

  const int steps = K >> 5;  // K-step 32

  // ---- async tile issue: global -> LDS, no VGPR staging ----
  auto issue = [&](int buf, int k0) {
#pragma unroll
    for (int i = 0; i < 8; ++i) {  // A tile: 64x32 floats, b64/lane
      int f = t + i * 128;
      int row = f >> 4, col = (f & 15) * 2;
      async_ld_b64(A + (size_t)(mBlock + row) * K + k0 + col,
                   &As[buf][row][col]);
    }
    if (!BT) {
#pragma unroll
      for (int i = 0; i < 4; ++i) {  // B tile: 32x64 floats, b128/lane
        int f = t + i * 128;
        int kr = f >> 4, c4 = (f & 15) * 4;
        async_ld_b128(B + (size_t)(k0 + kr) * N + nBlock + c4,
                      &Bsh[buf][kr * 72 + c4]);
      }
    } else {
#pragma unroll
      for (int i = 0; i < 8; ++i) {  // B tile: 64 n-rows x 32 k, b64/lane
        int f = t + i * 128;
        int nr = f >> 4, col = (f & 15) * 2;
        async_ld_b64(B + (size_t)(nBlock + nr) * K + k0 + col,
                     &Bsh[buf][nr * 36 + col]);
      }
    }
  };

  issue(0, 0);
  wait_async0();
  __syncthreads();

  for (int it = 0; it < steps; ++it) {
    const int cur = it & 1;
    if (it + 1 < steps) issue(cur ^ 1, (it + 1) * 32);

    // ---- 8 k-substeps of 4, 4 n-tiles -> 32 wmma per wave per K-step ----
#pragma unroll
    for (int kk = 0; kk < 8; ++kk) {
      const int kb = kk * 4;
      v2f a;
      a[0] = As[cur][mBase + lid][kb + 2 * lh + 0];
      a[1] = As[cur][mBase + lid][kb + 2 * lh + 1];
#pragma unroll
      for (int nt = 0; nt < 4; ++nt) {
        v2f b;
        if (!BT) {
          b[0] = Bsh[cur][(kb + 2 * lh + 0) * 72 + nt * 16 + lid];
          b[1] = Bsh[cur][(kb + 2 * lh + 1) * 72 + nt * 16 + lid];
        } else {
          b[0] = Bsh[cur][(nt * 16 + lid) * 36 + kb + 2 * lh + 0];
          b[1] = Bsh[cur][(nt * 16 + lid) * 36 + kb + 2 * lh + 1];
        }
        acc[nt] = __builtin_amdgcn_wmma_f32_16x16x4_f32(
            false, a, false, b, (short)0, acc[nt], false, false);
      }
    }

    wait_async0();   // prefetch issued this iteration has landed (own wave)
    __syncthreads(); // -> all waves' copies landed; safe to flip buffers
  }

  // ---- epilogue: C/D layout VGPR r -> rows r and r+8 ----
#pragma unroll
  for (int nt = 0; nt < 4; ++nt) {
#pragma unroll
    for (int r = 0; r < 8; ++r) {
      int row = mBlock + mBase + r + 8 * lh;
      int col = nBlock + nt * 16 + lid;
      float v = acc[nt][r];
      if (BIAS) v += bias[col];
      if (RESRELU) {
        v += Res[(size_t)row * N + col];
        v = fmaxf(v, 0.0f);
      }
      C[(size_t)row * N + col] = v;
    }
  }
}

// ---------------------------------------------------------------------------
// Power iteration helpers
// ---------------------------------------------------------------------------
__global__ void init_ones(float* v, int n, float val) {
  int i = blockIdx.x * blockDim.x + threadIdx.x;
  if (i < n) v[i] = val;
}

__global__ __launch_bounds__(256) void abs_matvec(const float* __restrict__ A,
                                                  const float* __restrict__ v,
                                                  float* __restrict__ w, int n) {
  __shared__ float red[256];
  const int row = blockIdx.x;
  float s = 0.0f;
  for (int j = threadIdx.x; j < n; j += 256)
    s += fabsf(A[(size_t)row * n + j]) * v[j];
  red[threadIdx.x] = s;
  __syncthreads();
  for (int off = 128; off > 0; off >>= 1) {
    if (threadIdx.x < off) red[threadIdx.x] += red[threadIdx.x + off];
    __syncthreads();
  }
  if (threadIdx.x == 0) w[row] = red[0];
}

__global__ __launch_bounds__(256) void l2norm(const float* __restrict__ w,
                                              float* __restrict__ out, int n,
                                              float addend) {
  __shared__ float red[256];
  float s = 0.0f;
  for (int j = threadIdx.x; j < n; j += 256) {
    float x = w[j];
    s += x * x;
  }
  red[threadIdx.x] = s;
  __syncthreads();
  for (int off = 128; off > 0; off >>= 1) {
    if (threadIdx.x < off) red[threadIdx.x] += red[threadIdx.x + off];
    __syncthreads();
  }
  if (threadIdx.x == 0) out[0] = sqrtf(red[0]) + addend;
}

__global__ void vec_scale(const float* __restrict__ w,
                          const float* __restrict__ nrm, float* __restrict__ v,
                          int n) {
  int i = blockIdx.x * blockDim.x + threadIdx.x;
  if (i < n) v[i] = w[i] / (nrm[0] + 1e-12f);
}

// ---------------------------------------------------------------------------
// projection_norm_inf: project each row of W (64x64) onto l1-ball radius
// vrad = kappa / rho.  One block of 64 threads, one row per thread.
// ---------------------------------------------------------------------------
__global__ __launch_bounds__(64) void project_rows(const float* __restrict__ W,
                                                   const float* __restrict__ rho,
                                                   float* __restrict__ Wp) {
  __shared__ float S[64][65];
  const int r = threadIdx.x;
  const float vrad = 0.9f / rho[0];
  float suma = 0.0f;
  for (int j = 0; j < 64; ++j) {
    float a = fabsf(W[r * 64 + j]);
    S[r][j] = a;
    suma += a;
  }
  // insertion sort descending (per-thread private LDS row)
  for (int i = 1; i < 64; ++i) {
    float key = S[r][i];
    int j = i - 1;
    while (j >= 0 && S[r][j] < key) {
      S[r][j + 1] = S[r][j];
      --j;
    }
    S[r][j + 1] = key;
  }
  float cs = 0.0f;
  int cnt = 0;
  for (int k = 0; k < 64; ++k) {
    cs += S[r][k];
    if (S[r][k] * (float)(k + 1) > cs - vrad) ++cnt;
  }
  float cs2 = 0.0f;
  for (int k = 0; k < cnt; ++k) cs2 += S[r][k];
  float theta = fmaxf((cs2 - vrad) / (float)cnt, 0.0f);
  const bool needs = suma > vrad;
  for (int j = 0; j < 64; ++j) {
    float wv = W[r * 64 + j];
    float pv = copysignf(fmaxf(fabsf(wv) - theta, 0.0f), wv);
    Wp[r * 64 + j] = needs ? pv : wv;
  }
}

// ---------------------------------------------------------------------------
// out[n, c] = sum_m X[m, n] * Vw[c, m]     (4096 x 16, K=64 -> tiny)
// ---------------------------------------------------------------------------
__global__ __launch_bounds__(256) void head_kernel(const float* __restrict__ X,
                                                   const float* __restrict__ Vw,
                                                   float* __restrict__ out) {
  int g = blockIdx.x * 256 + threadIdx.x;  // 0..65535
  int n = g >> 4, c = g & 15;
  float s = 0.0f;
#pragma unroll
  for (int m = 0; m < NHID; ++m) s += X[m * NNODE + n] * Vw[c * NHID + m];
  out[n * NCLASS + c] = s;
}

// ---------------------------------------------------------------------------
extern "C" void kernel_launch(void* const* d_in, const int* in_sizes, int n_in,
                              void* d_out, int out_size, void* d_ws,
                              size_t ws_size, hipStream_t stream) {
  (void)in_sizes; (void)n_in; (void)out_size; (void)ws_size;

  const float* features = (const float*)d_in[0];  // [128, 4096]
  const float* adj      = (const float*)d_in[1];  // [4096, 16384]
  const float* scaler_w = (const float*)d_in[2];  // [4096, 16384]
  const float* scaler_b = (const float*)d_in[3];  // [4096]
  const float* omega1   = (const float*)d_in[4];  // [64, 128]
  const float* W_ig     = (const float*)d_in[5];  // [64, 64]
  const float* V_w      = (const float*)d_in[6];  // [16, 64]
  const float* X0       = (const float*)d_in[7];  // [64, 4096]
  float* out = (float*)d_out;                     // [4096, 16]

  // workspace layout (floats)
  float* ws = (float*)d_ws;
  float* A_   = ws;                                   // 4096*4096
  float* Bmat = A_ + (size_t)NNODE * NNODE;           // 64*4096
  float* T    = Bmat + (size_t)NHID * NNODE;          // 64*4096
  float* Xa   = T + (size_t)NHID * NNODE;             // 64*4096
  float* Xb   = Xa + (size_t)NHID * NNODE;            // 64*4096
  float* vv   = Xb + (size_t)NHID * NNODE;            // 4096
  float* wv   = vv + NNODE;                           // 4096
  float* scal = wv + NNODE;                           // [0]=norm, [1]=rho
  float* Wp   = scal + 16;                            // 64*64

  // 1) A = adj @ scaler_w^T + b   (NT, bias)
  gemm_f32_wmma<true, true, false>
      <<<dim3(NNODE / 64, NNODE / 64), 128, 0, stream>>>(
          adj, scaler_w, scaler_b, nullptr, A_, NNODE, NNODE, 16384);

  // 2) power iteration on |A|
  init_ones<<<NNODE / 256, 256, 0, stream>>>(vv, NNODE, 1.0f / (float)NNODE);
  for (int it = 0; it < 50; ++it) {
    abs_matvec<<<NNODE, 256, 0, stream>>>(A_, vv, wv, NNODE);
    l2norm<<<1, 256, 0, stream>>>(wv, scal, NNODE, 0.0f);
    vec_scale<<<NNODE / 256, 256, 0, stream>>>(wv, scal, vv, NNODE);
  }
  abs_matvec<<<NNODE, 256, 0, stream>>>(A_, vv, wv, NNODE);
  l2norm<<<1, 256, 0, stream>>>(wv, scal + 1, NNODE, 1e-5f);  // rho

  // 3) Wp = projection_norm_inf(W_ig, kappa / rho)
  project_rows<<<1, 64, 0, stream>>>(W_ig, scal + 1, Wp);

  // 4) B = (omega1 @ features) @ A
  gemm_f32_wmma<false, false, false><<<dim3(NNODE / 64, 1), 128, 0, stream>>>(
      omega1, features, nullptr, nullptr, T, NHID, NNODE, 128);
  gemm_f32_wmma<false, false, false><<<dim3(NNODE / 64, 1), 128, 0, stream>>>(
      T, A_, nullptr, nullptr, Bmat, NHID, NNODE, NNODE);

  // 5) fixed point: X <- relu(Wp @ X @ A + B), 50 iters
  (void)hipMemcpyAsync(Xa, X0, (size_t)NHID * NNODE * sizeof(float),
                       hipMemcpyDeviceToDevice, stream);
  float* Xcur = Xa;
  float* Xnxt = Xb;
  for (int it = 0; it < 50; ++it) {
    gemm_f32_wmma<false, false, false><<<dim3(NNODE / 64, 1), 128, 0, stream>>>(
        Wp, Xcur, nullptr, nullptr, T, NHID, NNODE, NHID);
    gemm_f32_wmma<false, false, true><<<dim3(NNODE / 64, 1), 128, 0, stream>>>(
        T, A_, nullptr, Bmat, Xnxt, NHID, NNODE, NNODE);
    float* tmp = Xcur; Xcur = Xnxt; Xnxt = tmp;
  }

  // 6) out = X^T @ V^T
  head_kernel<<<(NNODE * NCLASS) / 256, 256, 0, stream>>>(Xcur, V_w, out);
}